// SimpleLSS_970662609519
// MI455X (gfx1250) — compile-verified
//
#include <hip/hip_runtime.h>
#include <hip/hip_bf16.h>
#include <math.h>

// ---------------------------------------------------------------------------
// SimpleLSS for MI455X (gfx1250, wave32).
//   weighted[n,c,h,w] = img[n,c,h,w] * S[n,h,w],  S = sum_d softmax(logits)_d
//   bev = bilinear_resize(weighted, 16x44 -> 128x128, half-pixel centers)
//
// Separable resize as two fp32 WMMA GEMMs per (n,c) plane:
//   Stage A: Tm(16x128)  = Wp(16x48)  @ Rw^T(48x128)
//   Stage B: Out(128x128) = Rh(128x16) @ Tm(16x128)
//
// The interpolation-weight fragments (Rw^T as B-frags, Rh as A-frags) are
// identical for all 6144 planes -> precomputed once into workspace in the
// exact V_WMMA_F32_16X16X4_F32 lane layout, L2-resident (32 KB), so the hot
// loop is LDS/L2 loads + WMMA only. Output (402 MB) is streamed with
// non-temporal stores; it is the bandwidth floor (~18 us @ 23.3 TB/s).
// ---------------------------------------------------------------------------

typedef __attribute__((ext_vector_type(2))) float v2f;
typedef __attribute__((ext_vector_type(8))) float v8f;

#define BN   24
#define CCH  256
#define DD   112
#define HH   16
#define WW   44
#define KPAD 48      // WW padded to multiple of 16 (stage-A K dimension)
#define OUTW 128

// Workspace layout (floats):
//   [0, 16896)                    : S[n,h,w] = sum_d softmax
//   [16896, 16896+6144)           : AW  stage-A B-fragments, 12*8*32 v2f
//   [23040, 23040+2048)           : BH  stage-B A-fragments,  4*8*32 v2f
#define S_FLOATS   (BN * HH * WW)          // 16896
#define AW_V2F     (12 * 8 * 32)           // 3072 v2f
#define BH_V2F     (4 * 8 * 32)            // 1024 v2f
#define AW_OFF     S_FLOATS
#define BH_OFF     (S_FLOATS + 2 * AW_V2F)

// ---------------------------------------------------------------------------
// Kernel 1: S[n,h,w] = sum_d softmax(logits[n,:,h,w])  (matches ref numerics:
// divide each term by E, then sum). 16896 positions -> trivial cost.
// ---------------------------------------------------------------------------
__global__ void lss_softmax_sum_kernel(const float* __restrict__ logits,
                                       float* __restrict__ S) {
  int idx = blockIdx.x * blockDim.x + threadIdx.x;
  if (idx >= BN * HH * WW) return;
  int n  = idx / (HH * WW);
  int hw = idx - n * (HH * WW);
  const float* p = logits + (size_t)n * DD * HH * WW + hw;
  float m = -INFINITY;
  for (int d = 0; d < DD; ++d) m = fmaxf(m, p[(size_t)d * HH * WW]);
  float e = 0.f;
  for (int d = 0; d < DD; ++d) e += __expf(p[(size_t)d * HH * WW] - m);
  float inv = 1.f / e;
  float s = 0.f;
  for (int d = 0; d < DD; ++d) s += __expf(p[(size_t)d * HH * WW] - m) * inv;
  S[idx] = s;
}

// Bilinear interpolation weight: contribution of input sample k to output x.
// Half-pixel centers, edge-clamped. Returns 0 for padded columns (k >= size).
__device__ __forceinline__ float rz_w(int x, int k, int in_size, float scale) {
  float src = ((float)x + 0.5f) * scale - 0.5f;
  float f   = floorf(src);
  float t   = src - f;
  int   i0  = (int)f;
  int   c0  = min(max(i0, 0), in_size - 1);
  int   c1  = min(max(i0 + 1, 0), in_size - 1);
  float w = 0.f;
  if (k == c0) w += 1.f - t;
  if (k == c1) w += t;
  return w;
}

// ---------------------------------------------------------------------------
// Kernel 1b: precompute WMMA weight fragments (shared by all 6144 planes).
//   AW[(ks*8+wv)*32+lane] = stage-A B-frag {Rw^T[kb][x], Rw^T[kb+1][x]}
//   BH[(ks*8+wv)*32+lane] = stage-B A-frag {Rh[y][kb],  Rh[y][kb+1]}
// Lane layout per ISA: lanes 0-15 carry K = {kb, kb+1}, lanes 16-31 {kb+2,kb+3}.
// ---------------------------------------------------------------------------
__global__ void lss_weights_kernel(float* __restrict__ ws) {
  v2f* AW = (v2f*)(ws + AW_OFF);
  v2f* BH = (v2f*)(ws + BH_OFF);
  const float scaleW = (float)WW / (float)OUTW;
  const float scaleH = (float)HH / (float)OUTW;
  int i = blockIdx.x * blockDim.x + threadIdx.x;
  if (i < AW_V2F) {
    int ks = i / (8 * 32), r = i % (8 * 32);
    int wv = r / 32, lane = r % 32;
    int kb = ks * 4 + ((lane & 16) ? 2 : 0);
    int x  = wv * 16 + (lane & 15);
    v2f v;
    v.x = rz_w(x, kb,     WW, scaleW);
    v.y = rz_w(x, kb + 1, WW, scaleW);
    AW[i] = v;
  } else if (i < AW_V2F + BH_V2F) {
    int j  = i - AW_V2F;
    int ks = j / (8 * 32), r = j % (8 * 32);
    int wv = r / 32, lane = r % 32;
    int kb = ks * 4 + ((lane & 16) ? 2 : 0);
    int y  = wv * 16 + (lane & 15);
    v2f v;
    v.x = rz_w(y, kb,     HH, scaleH);
    v.y = rz_w(y, kb + 1, HH, scaleH);
    BH[j] = v;
  }
}

// ---------------------------------------------------------------------------
// Kernel 2: one 256-thread block (8 wave32) per (n,c) plane.
// ---------------------------------------------------------------------------
__global__ __launch_bounds__(256) void lss_wmma_resize_kernel(
    const float* __restrict__ img, const float* __restrict__ S,
    const v2f* __restrict__ AW, const v2f* __restrict__ BH,
    float* __restrict__ out) {
  __shared__ float Wp[HH][KPAD];   // depth-weighted plane, zero-padded K
  __shared__ float Tm[HH][OUTW];   // stage-A result (K x N for stage B)

  const int plane = blockIdx.x;          // n*256 + c
  const int n     = plane >> 8;
  const int c     = plane & (CCH - 1);

  const int lane  = threadIdx.x & 31;
  const int wv    = threadIdx.x >> 5;    // wave id 0..7
  const bool hi   = lane >= 16;          // upper half-wave (K+2/K+3, M+8 rows)
  const int l15   = lane & 15;

  // ---- load plane (single-use -> non-temporal), apply S, pad K to 48 ----
  const float* imgp = img + (((size_t)n * CCH + c) * HH) * WW;
  const float* Sp   = S + (size_t)n * HH * WW;
  for (int i = threadIdx.x; i < HH * KPAD; i += 256) {
    int h = i / KPAD, w = i - h * KPAD;
    float v = 0.f;
    if (w < WW) v = __builtin_nontemporal_load(&imgp[h * WW + w]) * Sp[h * WW + w];
    Wp[h][w] = v;
  }
  __syncthreads();

  // ---- Stage A: Tm = Wp @ Rw^T ; wave wv owns columns [wv*16, wv*16+16) ----
  {
    const int xcol = wv * 16 + l15;      // output x this lane contributes to
    v8f acc = {};
#pragma unroll
    for (int ks = 0; ks < KPAD / 4; ++ks) {
      const int kb = ks * 4 + (hi ? 2 : 0);
      v2f a;
      a.x = Wp[l15][kb];                 // A: M=h rows, K=w_in
      a.y = Wp[l15][kb + 1];
      const v2f b = AW[(ks * 8 + wv) * 32 + lane];   // precomputed B-frag
      acc = __builtin_amdgcn_wmma_f32_16x16x4_f32(
          /*neg_a=*/false, a, /*neg_b=*/false, b,
          /*c_mod=*/(short)0, acc, /*reuse_a=*/false, /*reuse_b=*/false);
    }
    // scatter D tile (16x16 f32 layout) into LDS Tm
    const int r0 = hi ? 8 : 0;
#pragma unroll
    for (int r = 0; r < 8; ++r) Tm[r0 + r][xcol] = acc[r];
  }
  __syncthreads();

  // ---- Stage B: Out = Rh @ Tm ; wave wv owns rows [wv*16, wv*16+16) ----
  {
    // A-fragments (Rh) are invariant across the 8 column tiles: load once.
    v2f ah[HH / 4];
#pragma unroll
    for (int ks = 0; ks < HH / 4; ++ks)
      ah[ks] = BH[(ks * 8 + wv) * 32 + lane];

    const int ybase = wv * 16;
    float* outp = out + (size_t)plane * OUTW * OUTW;
    for (int nt = 0; nt < OUTW / 16; ++nt) {
      const int xb = nt * 16 + l15;
      v8f acc = {};
#pragma unroll
      for (int ks = 0; ks < HH / 4; ++ks) {
        const int kb = ks * 4 + (hi ? 2 : 0);
        v2f b;
        b.x = Tm[kb][xb];                      // B: K=h_in, N=x_out
        b.y = Tm[kb + 1][xb];
        acc = __builtin_amdgcn_wmma_f32_16x16x4_f32(
            false, ah[ks], false, b, (short)0, acc, false, false);
      }
      // store D tile non-temporally: VGPR r -> row ybase+(hi?8:0)+r, col xb
      const size_t base = (size_t)(ybase + (hi ? 8 : 0)) * OUTW + nt * 16 + l15;
#pragma unroll
      for (int r = 0; r < 8; ++r)
        __builtin_nontemporal_store(acc[r], &outp[base + (size_t)r * OUTW]);
    }
  }
}

extern "C" void kernel_launch(void* const* d_in, const int* in_sizes, int n_in,
                              void* d_out, int out_size, void* d_ws, size_t ws_size,
                              hipStream_t stream) {
  (void)in_sizes; (void)n_in; (void)out_size; (void)ws_size;
  const float* img    = (const float*)d_in[0];   // (24,256,16,44) f32
  const float* logits = (const float*)d_in[1];   // (24,112,16,44) f32
  float*       out    = (float*)d_out;           // (24,256,128,128) f32
  float*       ws     = (float*)d_ws;            // S + weight fragments

  const int npos = BN * HH * WW;
  lss_softmax_sum_kernel<<<(npos + 255) / 256, 256, 0, stream>>>(logits, ws);

  const int nw = AW_V2F + BH_V2F;
  lss_weights_kernel<<<(nw + 255) / 256, 256, 0, stream>>>(ws);

  lss_wmma_resize_kernel<<<BN * CCH, 256, 0, stream>>>(
      img, ws, (const v2f*)(ws + AW_OFF), (const v2f*)(ws + BH_OFF), out);
}